// YOLOLoss_43533788512442
// MI455X (gfx1250) — compile-verified
//
#include <hip/hip_runtime.h>
#include <hip/hip_bf16.h>
#include <math.h>

#define LAMBDA_COORD 5.0f
#define LAMBDA_NOOBJ 0.5f
#define BB 64
#define AA 10647
#define CC 9
#define GG 50
#define ROW 14          // floats per pred row (56 bytes)
#define TILE 256
#define GT_PER_BLOCK 8

typedef float v2f __attribute__((ext_vector_type(2)));
typedef float v8f __attribute__((ext_vector_type(8)));

__device__ __forceinline__ float clamp_log(float x) {
    return fmaxf(logf(x), -100.0f);
}

// ---------------------------------------------------------------------------
// Kernel 1: tot0[b] = sum_a -clamp_log(1 - conf[b][a])
// ---------------------------------------------------------------------------
__global__ void yolo_tot0_kernel(const float* __restrict__ pred,
                                 float* __restrict__ tot0) {
    const int b   = blockIdx.x;
    const int tid = threadIdx.x;
    const float* base = pred + (size_t)b * AA * ROW;
    float s = 0.0f;
    for (int a = tid; a < AA; a += 256) {
        float conf = base[a * ROW + 4];
        s -= fmaxf(logf(1.0f - conf), -100.0f);
    }
    __shared__ float red[256];
    red[tid] = s;
    __syncthreads();
    for (int o = 128; o > 0; o >>= 1) {
        if (tid < o) red[tid] += red[tid + o];
        __syncthreads();
    }
    if (tid == 0) tot0[b] = red[0];
}

// ---------------------------------------------------------------------------
// Kernel 2: best[b][g] = argmax_a IoU(pred_box[b][a], gt[b][g])
// Anchor boxes staged to LDS with CDNA5 async global->LDS DMA, double-buffered.
// Each of the 8 waves owns one GT and scans the shared tile.
// ---------------------------------------------------------------------------
__global__ void yolo_argmax_kernel(const float* __restrict__ pred,
                                   const float* __restrict__ bboxes,
                                   int* __restrict__ best) {
    const int b     = blockIdx.y;
    const int gBase = blockIdx.x * GT_PER_BLOCK;
    const int tid   = threadIdx.x;
    const int lane  = tid & 31;
    const int wave  = tid >> 5;
    const int g     = gBase + wave;
    const bool active = (g < GG);

    __shared__ float4 buf[2][TILE];

    const unsigned long long base =
        (unsigned long long)(pred + (size_t)b * AA * ROW);

    // GT box: corner form re-derived from center form, mirroring the reference
    float gx1 = 0.f, gx2 = 0.f, gy1 = 0.f, gy2 = 0.f, ag = 0.f;
    if (active) {
        const float* gb = bboxes + ((size_t)b * GG + g) * 4;
        float bx1 = gb[0], by1 = gb[1], bx2 = gb[2], by2 = gb[3];
        float gcx = (bx1 + bx2) * 0.5f, gcy = (by1 + by2) * 0.5f;
        float gw  = bx2 - bx1,          gh  = by2 - by1;
        gx1 = gcx - gw * 0.5f; gx2 = gcx + gw * 0.5f;
        gy1 = gcy - gh * 0.5f; gy2 = gcy + gh * 0.5f;
        ag  = (gx2 - gx1) * (gy2 - gy1);
    }

    const int nTiles = (AA + TILE - 1) / TILE;

    // Prologue: async-stage tile 0 (first 16 bytes of each 56-byte row = box)
    {
        int a = tid; if (a > AA - 1) a = AA - 1;           // clamp tail (dup ok)
        unsigned vOff   = (unsigned)(a * (ROW * 4));
        unsigned ldsOff = (unsigned)(size_t)(&buf[0][tid]);
        asm volatile(
            "global_load_async_to_lds_b64 %0, %1, %2\n\t"
            "global_load_async_to_lds_b64 %0, %1, %2 offset:8"
            :: "v"(ldsOff), "v"(vOff), "s"(base) : "memory");
    }

    float bestIou = -1.0f;
    int   bestIdx = 0;

    for (int t = 0; t < nTiles; ++t) {
        if (t + 1 < nTiles) {
            int a = (t + 1) * TILE + tid; if (a > AA - 1) a = AA - 1;
            unsigned vOff   = (unsigned)(a * (ROW * 4));
            unsigned ldsOff = (unsigned)(size_t)(&buf[(t + 1) & 1][tid]);
            asm volatile(
                "global_load_async_to_lds_b64 %0, %1, %2\n\t"
                "global_load_async_to_lds_b64 %0, %1, %2 offset:8"
                :: "v"(ldsOff), "v"(vOff), "s"(base) : "memory");
            // 2 ops just issued for t+1 remain; tile t's 2 ops must retire
            asm volatile("s_wait_asynccnt 0x2" ::: "memory");
        } else {
            asm volatile("s_wait_asynccnt 0x0" ::: "memory");
        }
        __syncthreads();   // all waves' tile-t data now resident in LDS

        if (active) {
            const int tBase = t * TILE;
            const float4* cur = buf[t & 1];
            for (int j = lane; j < TILE; j += 32) {
                int a = tBase + j;
                if (a < AA) {
                    float4 p  = cur[j];
                    float phw = p.z * 0.5f, phh = p.w * 0.5f;
                    float px1 = p.x - phw, px2 = p.x + phw;
                    float py1 = p.y - phh, py2 = p.y + phh;
                    float iw  = fmaxf(fminf(px2, gx2) - fmaxf(px1, gx1), 0.0f);
                    float ih  = fmaxf(fminf(py2, gy2) - fmaxf(py1, gy1), 0.0f);
                    float inter = iw * ih;
                    float apb = (px2 - px1) * (py2 - py1);
                    float iou = inter / (apb + ag - inter + 1e-16f);
                    if (iou > bestIou) { bestIou = iou; bestIdx = a; }
                }
            }
        }
        __syncthreads();   // everyone done reading buf[t&1] before it is reused
    }

    if (active) {
        // wave reduction: max IoU, ties -> smallest index (argmax-first)
        for (int off = 16; off >= 1; off >>= 1) {
            float oi = __shfl_down(bestIou, off, 32);
            int   ox = __shfl_down(bestIdx, off, 32);
            if (oi > bestIou || (oi == bestIou && ox < bestIdx)) {
                bestIou = oi; bestIdx = ox;
            }
        }
        if (lane == 0) best[b * GG + g] = bestIdx;
    }
}

// ---------------------------------------------------------------------------
// Kernel 3: per-image loss assembly + WMMA-based 64-way batch reduction
// ---------------------------------------------------------------------------
__global__ void yolo_final_kernel(const float* __restrict__ pred,
                                  const float* __restrict__ bboxes,
                                  const int* __restrict__ classes,
                                  const float* __restrict__ tot0,
                                  const int* __restrict__ best,
                                  float* __restrict__ out) {
    __shared__ float s_per[BB];
    const int tid = threadIdx.x;   // 64 threads = 2 waves
    if (tid < BB) {
        const int b = tid;
        const float t0 = tot0[b];
        float per = 0.0f;
        bool  has = false;
        for (int g = 0; g < GG; ++g) {
            int cls = classes[b * GG + g];
            if (cls == -1) continue;
            has = true;
            int a = best[b * GG + g];
            const float* row = pred + ((size_t)b * AA + a) * ROW;
            const float* gb  = bboxes + ((size_t)b * GG + g) * 4;
            float bx1 = gb[0], by1 = gb[1], bx2 = gb[2], by2 = gb[3];
            float gcx = (bx1 + bx2) * 0.5f, gcy = (by1 + by2) * 0.5f;
            float gw  = bx2 - bx1,          gh  = by2 - by1;
            float dx = row[0] - gcx, dy = row[1] - gcy;
            float dw = row[2] - gw,  dh = row[3] - gh;
            float coord = LAMBDA_COORD * (dx*dx + dy*dy + dw*dw + dh*dh);
            float conf      = row[4];
            float conf_obj  = -clamp_log(conf);
            float cls_loss  = 0.0f;
            #pragma unroll
            for (int c = 0; c < CC; ++c) {
                float p = row[5 + c];
                cls_loss += (c == cls) ? -clamp_log(p) : -clamp_log(1.0f - p);
            }
            float bce0_best = -clamp_log(1.0f - conf);
            float noobj = LAMBDA_NOOBJ * (t0 - bce0_best);
            per += coord + conf_obj + cls_loss + noobj;
        }
        if (!has) per = LAMBDA_NOOBJ * t0;
        s_per[b] = per;
    }
    __syncthreads();

    // Wave 0: reduce the 64 per-image values with one v_wmma_f32_16x16x4_f32.
    // A = 16x4 matrix of the values, B = ones => D[m][n] = rowsum_m.
    // Column N=0 lives in lane 0 (M=0..7) and lane 16 (M=8..15).
    if (tid < 32) {
        int m  = tid & 15;
        int kh = tid >> 4;                 // 0: K=0,1   1: K=2,3
        v2f a; a[0] = s_per[m * 4 + 2 * kh];
               a[1] = s_per[m * 4 + 2 * kh + 1];
        v2f bone; bone[0] = 1.0f; bone[1] = 1.0f;
        v8f c = {};
        v8f d = __builtin_amdgcn_wmma_f32_16x16x4_f32(
            /*neg_a=*/false, a, /*neg_b=*/false, bone,
            /*c_mod=*/(short)0, c, /*reuse_a=*/false, /*reuse_b=*/false);
        float tsum = 0.0f;
        #pragma unroll
        for (int i = 0; i < 8; ++i) tsum += d[i];
        float other = __shfl(tsum, 16, 32);   // lane16 carries rows 8..15
        if (tid == 0) out[0] = (tsum + other) * (1.0f / (float)BB);
    }
}

// ---------------------------------------------------------------------------
extern "C" void kernel_launch(void* const* d_in, const int* in_sizes, int n_in,
                              void* d_out, int out_size, void* d_ws, size_t ws_size,
                              hipStream_t stream) {
    const float* pred    = (const float*)d_in[0];   // (B, A, 14) f32
    const float* bboxes  = (const float*)d_in[1];   // (B, G, 4)  f32
    const int*   classes = (const int*)d_in[2];     // (B, G)     i32
    float* out = (float*)d_out;

    float* tot0 = (float*)d_ws;                         // 64 floats
    int*   best = (int*)((char*)d_ws + 256);            // 3200 ints

    yolo_tot0_kernel<<<BB, 256, 0, stream>>>(pred, tot0);
    yolo_argmax_kernel<<<dim3((GG + GT_PER_BLOCK - 1) / GT_PER_BLOCK, BB),
                         256, 0, stream>>>(pred, bboxes, best);
    yolo_final_kernel<<<1, 64, 0, stream>>>(pred, bboxes, classes, tot0, best, out);
}